// NSAMSA_33870112096812
// MI455X (gfx1250) — compile-verified
//
#include <hip/hip_runtime.h>

// ---------------------------------------------------------------------------
// NSA-style ball attention forward for MI455X (gfx1250, wave32, WMMA).
//
// Shapes: D=128, H=8 heads, E=16 head dim, N=128 balls, M=64 pts/ball,
//         NM=8192 points, TOPK=2, window = 128 keys.
//
// Forward-math simplifications:
//   * straight-through gates == 1.0 in forward  -> no gating
//   * selection softmax is monotonic            -> top-2 of raw q.keys logits
//   * mean_m(rel) == 0                          -> keys = mean_m(x)
// ---------------------------------------------------------------------------

typedef float v2f __attribute__((ext_vector_type(2)));
typedef float v8f __attribute__((ext_vector_type(8)));

#define DD   128   // model dim
#define HH   8     // heads
#define MM   64    // ball size
#define NN   128   // num balls
#define EE   16    // head dim
#define NMNM 8192  // total points
#define WIN  128   // TOPK * MM

// ---------------------------------------------------------------------------
// Kernel A: xr = x + (pos - mean_m pos);  keys[n][d] = mean_m x[n,m,d]
// One block per ball (n), 128 threads = one per model dim d. Coalesced.
// ---------------------------------------------------------------------------
__global__ void nsa_prep(const float* __restrict__ x, const float* __restrict__ pos,
                         float* __restrict__ xr, float* __restrict__ keys) {
    const int n = blockIdx.x;
    const int d = threadIdx.x;
    const int base = n * MM * DD + d;

    float sp = 0.f, sx = 0.f;
    for (int m = 0; m < MM; ++m) {
        sp += pos[base + m * DD];
        sx += x[base + m * DD];
    }
    const float mp = sp * (1.0f / MM);
    keys[n * DD + d] = sx * (1.0f / MM);

    for (int m = 0; m < MM; ++m) {
        xr[base + m * DD] = x[base + m * DD] + pos[base + m * DD] - mp;
    }
}

// ---------------------------------------------------------------------------
// Kernel B: ball selection. One wave per (head h, tile of 16 points).
// sim tile (16 pts x 16 balls) = Q(16x16) x KeysT(16x16) via 4x
// v_wmma_f32_16x16x4_f32; 8 ball-tiles cover all 128 balls. Tile spilled to
// LDS (row-padded to kill bank conflicts), then both half-waves scan 64
// logits for their point and merge top-2 across halves via shuffles.
// ---------------------------------------------------------------------------
__global__ void nsa_select(const float* __restrict__ xr, const float* __restrict__ keys,
                           int* __restrict__ top2) {
    __shared__ float simbuf[16][NN + 1];     // +1 pad: rows land on distinct banks

    const int h  = blockIdx.x >> 9;          // 512 point-tiles per head
    const int p0 = (blockIdx.x & 511) << 4;  // first point of the tile
    const int l  = threadIdx.x;              // lane 0..31
    const int lm = l & 15;
    const int lh = (l >> 4) << 1;            // 0 (lanes 0-15) or 2 (lanes 16-31)

    // A fragments: Q tile rows = points, cols = head dims.
    // 32-bit A 16x4 layout: lane = row M, VGPR0/1 = K pair {lh, lh+1}.
    v2f afrag[4];
    const int qrow = p0 + lm;
    #pragma unroll
    for (int kk = 0; kk < 4; ++kk) {
        const int c0 = h * EE + kk * 4 + lh;
        afrag[kk].x = xr[qrow * DD + c0];
        afrag[kk].y = xr[qrow * DD + c0 + 1];
    }

    for (int nb = 0; nb < 8; ++nb) {
        v8f acc = {};
        const int ball = nb * 16 + lm;
        #pragma unroll
        for (int kk = 0; kk < 4; ++kk) {
            // B fragment: B[k][n] = keys[ball n][e = kk*4 + k]; row striping
            // mirrors C/D: VGPR0 rows {lh}, VGPR1 rows {lh+1}, col = lane%16.
            v2f b;
            const int c0 = h * EE + kk * 4 + lh;
            b.x = keys[ball * DD + c0];
            b.y = keys[ball * DD + c0 + 1];
            acc = __builtin_amdgcn_wmma_f32_16x16x4_f32(false, afrag[kk], false, b,
                                                        (short)0, acc, false, false);
        }
        // D layout: VGPR v holds row v (lanes 0-15) / v+8 (lanes 16-31), col l%16.
        const int rbase = (l >> 4) << 3;
        #pragma unroll
        for (int v = 0; v < 8; ++v)
            simbuf[rbase + v][nb * 16 + lm] = acc[v];
    }
    __syncthreads();

    // Parallel top-2: each half-wave scans 64 logits of point `lm`, then merge.
    // Strict '>' keeps the lower index on ties (matches jax top_k); low-half
    // indices are always smaller, so the cross-half merge tie-break is exact.
    const int jb = (l >> 4) * 64;
    float best = -3.4e38f, sec = -3.4e38f;
    int bi = 0, si = 0;
    for (int j = 0; j < 64; ++j) {
        const float v = simbuf[lm][jb + j];
        if (v > best)     { sec = best; si = bi; best = v; bi = jb + j; }
        else if (v > sec) { sec = v;   si = jb + j; }
    }
    const float ob = __shfl_xor(best, 16, 32);
    const int   oi = __shfl_xor(bi,   16, 32);
    const float os = __shfl_xor(sec,  16, 32);
    const int   oj = __shfl_xor(si,   16, 32);
    if (l < 16) {            // this lane holds the LOW half; other = high half
        int rb, rs;
        if (ob > best) { rb = oi; rs = (os > best) ? oj : bi; }
        else           { rb = bi; rs = (ob > sec)  ? oi : si; }
        const int p = p0 + lm;
        top2[(p * HH + h) * 2 + 0] = rb;
        top2[(p * HH + h) * 2 + 1] = rs;
    }
}

// ---------------------------------------------------------------------------
// Kernel C: windowed SDPA. One block per point (8 waves = 8 heads).
// Logits: 8 key-tiles, each D = K(16x16) x B where B broadcasts q across all
// 16 columns (every column of D equals K.q -> no masking needed). While
// feeding the WMMAs, lanes stage their K elements into LDS (K == V), so the
// A.V reduction runs entirely out of LDS. Softmax max/sum use a 5-step
// shuffle butterfly instead of serial 128-element scans.
// ---------------------------------------------------------------------------
__global__ void nsa_attend(const float* __restrict__ xr, const int* __restrict__ top2,
                           float* __restrict__ out) {
    __shared__ float kv[HH][WIN][EE + 1];    // staged K/V window, row-padded
    __shared__ float s_lds[HH][WIN];         // logits -> attention weights

    const int p  = blockIdx.x;
    const int h  = threadIdx.x >> 5;   // wave index = head
    const int l  = threadIdx.x & 31;
    const int lm = l & 15;
    const int lh = (l >> 4) << 1;

    const int i0 = top2[(p * HH + h) * 2 + 0];
    const int i1 = top2[(p * HH + h) * 2 + 1];

    // Warm the L2/L0 path for the two selected balls (global_prefetch_b8).
    __builtin_prefetch(&xr[(size_t)i0 * MM * DD + h * EE], 0, 3);
    __builtin_prefetch(&xr[(size_t)i1 * MM * DD + h * EE], 0, 3);

    // B fragments: q broadcast -> every column of B identical.
    v2f bfrag[4];
    #pragma unroll
    for (int kk = 0; kk < 4; ++kk) {
        const int c0 = h * EE + kk * 4 + lh;
        bfrag[kk].x = xr[p * DD + c0];
        bfrag[kk].y = xr[p * DD + c0 + 1];
    }

    for (int t = 0; t < 8; ++t) {
        const int wrow = t * 16 + lm;                 // window row for this lane
        const int ball = (wrow < MM) ? i0 : i1;
        const int g    = ball * MM + (wrow & (MM - 1));
        v8f acc = {};
        #pragma unroll
        for (int kk = 0; kk < 4; ++kk) {
            const int c0 = kk * 4 + lh;
            v2f a;
            a.x = xr[g * DD + h * EE + c0];
            a.y = xr[g * DD + h * EE + c0 + 1];
            kv[h][wrow][c0]     = a.x;               // stage K(=V) for the AV pass
            kv[h][wrow][c0 + 1] = a.y;
            acc = __builtin_amdgcn_wmma_f32_16x16x4_f32(false, a, false, bfrag[kk],
                                                        (short)0, acc, false, false);
        }
        if (lm == 0) {  // lanes 0 (rows 0-7) and 16 (rows 8-15): col 0 of D
            const int rbase = (l >> 4) << 3;
            #pragma unroll
            for (int v = 0; v < 8; ++v)
                s_lds[h][t * 16 + rbase + v] = acc[v] * 0.25f;  // ATT_SCALE = E^-0.5
        }
    }
    __syncthreads();

    // Softmax over 128 logits: 4 elements/lane + shuffle-butterfly reductions.
    float sj[4];
    #pragma unroll
    for (int u = 0; u < 4; ++u) sj[u] = s_lds[h][l * 4 + u];

    float mx = fmaxf(fmaxf(sj[0], sj[1]), fmaxf(sj[2], sj[3]));
    #pragma unroll
    for (int off = 16; off > 0; off >>= 1)
        mx = fmaxf(mx, __shfl_xor(mx, off, 32));

    float ej[4], se = 0.f;
    #pragma unroll
    for (int u = 0; u < 4; ++u) { ej[u] = __expf(sj[u] - mx); se += ej[u]; }
    #pragma unroll
    for (int off = 16; off > 0; off >>= 1)
        se += __shfl_xor(se, off, 32);
    const float inv = 1.0f / se;

    // Overwrite logits with normalized weights (same-wave LDS is in-order).
    #pragma unroll
    for (int u = 0; u < 4; ++u)
        s_lds[h][l * 4 + u] = ej[u] * inv;

    // out[e] = sum_j att[j] * V[j][e]; half-waves split the 128-key window,
    // all operands from LDS.
    const int e    = lm;
    const int jbase = (l >> 4) * MM;
    float acc = 0.f;
    for (int jj = 0; jj < MM; ++jj)
        acc += s_lds[h][jbase + jj] * kv[h][jbase + jj][e];
    acc += __shfl_xor(acc, 16, 32);
    if (l < 16)
        out[p * DD + h * EE + e] = acc;   // out[p, h, e] (H*E == D)
}

// ---------------------------------------------------------------------------
extern "C" void kernel_launch(void* const* d_in, const int* in_sizes, int n_in,
                              void* d_out, int out_size, void* d_ws, size_t ws_size,
                              hipStream_t stream) {
    const float* x   = (const float*)d_in[0];
    const float* pos = (const float*)d_in[1];
    float*       out = (float*)d_out;

    // workspace layout (~4.8 MB): xr | keys | top2
    float* xr   = (float*)d_ws;
    float* keys = xr + (size_t)NMNM * DD;
    int*   top2 = (int*)(keys + (size_t)NN * DD);

    nsa_prep  <<<NN,               DD,      0, stream>>>(x, pos, xr, keys);
    nsa_select<<<HH * (NMNM / 16), 32,      0, stream>>>(xr, keys, top2);
    nsa_attend<<<NMNM,             HH * 32, 0, stream>>>(xr, top2, out);
}